// MTKDecoderSelfAttentionKVCacheV2_18957985644735
// MI455X (gfx1250) — compile-verified
//
#include <hip/hip_runtime.h>
#include <math.h>

#define D_MODEL   2048
#define NUM_HEADS 16
#define HEAD_DIM  128
#define BATCH     16
#define MAX_CACHE 4096
#define SEQ       (MAX_CACHE + 1)            // 4097
#define NTILE     (D_MODEL / 16)             // 128 column tiles
#define KSPLIT    16
#define KRANGE    (D_MODEL / KSPLIT)         // 128
#define CHUNK     256                        // keys per block (4 waves x 64)
#define NCHUNK    ((SEQ + CHUNK - 1) / CHUNK)// 17
#define PART_STRIDE 132                      // 128 acc + m + l + pad
#define SCALING   0.08838834764831845f       // 128^-0.5

typedef __attribute__((ext_vector_type(2))) float v2f;
typedef __attribute__((ext_vector_type(4))) float f4;
typedef __attribute__((ext_vector_type(8))) float v8f;

static __device__ __forceinline__ float dot4(f4 a, f4 b) {
    return a.x * b.x + a.y * b.y + a.z * b.z + a.w * b.w;
}
static __device__ __forceinline__ float wave_allreduce_sum(float p) {
    p += __shfl_xor(p, 16, 32);
    p += __shfl_xor(p, 8, 32);
    p += __shfl_xor(p, 4, 32);
    p += __shfl_xor(p, 2, 32);
    p += __shfl_xor(p, 1, 32);
    return p;
}

// ---------------------------------------------------------------------------
// Partial GEMM: D_partial = X[16, k-range] @ W[., k-range]^T for one 16x16
// tile and one K-split. grid.x = (proj*NTILE + ntile)*KSPLIT + ks.
// One wave per block; V_WMMA_F32_16X16X4_F32 accumulation, fp32 throughout.
// ---------------------------------------------------------------------------
__global__ __launch_bounds__(32)
void gemm16_wmma_partial(const float* __restrict__ X,
                         const float* __restrict__ W0,
                         const float* __restrict__ W1,
                         const float* __restrict__ W2,
                         float* __restrict__ part)   // [grid.x][256]
{
    const int lane  = threadIdx.x;
    const int idx   = blockIdx.x;
    const int ks    = idx % KSPLIT;
    const int ntile = (idx / KSPLIT) % NTILE;
    const int proj  = idx / (KSPLIT * NTILE);
    const float* W  = (proj == 0) ? W0 : (proj == 1) ? W1 : W2;

    const int n0   = ntile * 16;
    const int mrow = lane & 15;
    const int koff = (lane >> 4) * 2 + ks * KRANGE;

    const float* xp = X + mrow * D_MODEL + koff;
    const float* wp = W + (size_t)(n0 + mrow) * D_MODEL + koff;

    v8f c = {};
#pragma unroll 8
    for (int k = 0; k < KRANGE; k += 4) {
        v2f a = *(const v2f*)(xp + k);
        v2f b = *(const v2f*)(wp + k);
        c = __builtin_amdgcn_wmma_f32_16x16x4_f32(
                false, a, false, b, (short)0, c, false, false);
    }
    *(v8f*)(part + (size_t)idx * 256 + lane * 8) = c;
}

// ---------------------------------------------------------------------------
// Reduce KSPLIT partials of each 16x16 tile, add bias, scale, store.
// grid.x = proj*NTILE + ntile.
// ---------------------------------------------------------------------------
__global__ __launch_bounds__(32)
void gemm16_reduce(const float* __restrict__ part,
                   const float* __restrict__ b0,
                   const float* __restrict__ b1,
                   const float* __restrict__ b2,
                   float* __restrict__ d0,
                   float* __restrict__ d1,
                   float* __restrict__ d2,
                   float scale0)
{
    const int lane  = threadIdx.x;
    const int ntile = blockIdx.x % NTILE;
    const int proj  = blockIdx.x / NTILE;
    const float* bias = (proj == 0) ? b0 : (proj == 1) ? b1 : b2;
    float*       D    = (proj == 0) ? d0 : (proj == 1) ? d1 : d2;
    const float scale = (proj == 0) ? scale0 : 1.0f;

    const float* pb = part + (size_t)blockIdx.x * KSPLIT * 256 + lane * 8;
    v8f c = {};
#pragma unroll
    for (int ks = 0; ks < KSPLIT; ++ks) {
        v8f p = *(const v8f*)(pb + ks * 256);
        c = c + p;
    }
    const int col   = ntile * 16 + (lane & 15);
    const int mbase = (lane >> 4) * 8;
    const float bv  = bias[col];
#pragma unroll
    for (int v = 0; v < 8; ++v)
        D[(mbase + v) * D_MODEL + col] = (c[v] + bv) * scale;
}

// ---------------------------------------------------------------------------
// Flash-decode partial: block = 4 waves, CHUNK=256 keys of one (b,h).
// Wave handles 64 keys; lane owns dims [4L,4L+4) -> coalesced 512B row loads.
// 2-key unroll: more loads in flight, 3 exps per 2 keys. Waves merge their
// online-softmax states in LDS; one (m,l,acc[128]) partial per block.
// ---------------------------------------------------------------------------
__global__ __launch_bounds__(128)
void attn_partial(const float* __restrict__ pastK,
                  const float* __restrict__ pastV,
                  const float* __restrict__ newK,
                  const float* __restrict__ newV,
                  const float* __restrict__ mask,
                  const float* __restrict__ qs,
                  float* __restrict__ part)  // [B*H, NCHUNK, PART_STRIDE]
{
    __shared__ float smem[4 * PART_STRIDE];

    const int tid  = threadIdx.x;
    const int wave = tid >> 5;
    const int lane = tid & 31;
    const int dof  = lane * 4;

    const int bid   = blockIdx.x;
    const int chunk = bid % NCHUNK;
    const int bh    = bid / NCHUNK;
    const int h     = bh % NUM_HEADS;
    const int b     = bh / NUM_HEADS;

    const size_t headoff = (size_t)h * HEAD_DIM + dof;
    const f4 qv = *(const f4*)(qs + (size_t)b * D_MODEL + headoff);

    const int s0 = chunk * CHUNK + wave * 64;
    int s1 = s0 + 64;
    if (s1 > SEQ) s1 = SEQ;
    const int sPastEnd = (s1 < MAX_CACHE) ? s1 : MAX_CACHE;  // even count from s0

    float m = -INFINITY, l = 0.0f;
    f4 acc = {0.0f, 0.0f, 0.0f, 0.0f};

    const float* kbase = pastK + (size_t)b * MAX_CACHE * D_MODEL + headoff;
    const float* vbase = pastV + (size_t)b * MAX_CACHE * D_MODEL + headoff;

    int s = s0;
    for (; s + 1 < sPastEnd; s += 2) {
        const float* kp = kbase + (size_t)s * D_MODEL;
        const float* vp = vbase + (size_t)s * D_MODEL;
        f4 k0 = *(const f4*)kp;
        f4 k1 = *(const f4*)(kp + D_MODEL);
        f4 v0 = *(const f4*)vp;
        f4 v1 = *(const f4*)(vp + D_MODEL);

        float p0 = wave_allreduce_sum(dot4(qv, k0));
        float p1 = wave_allreduce_sum(dot4(qv, k1));
        float sc0 = p0 + mask[s];
        float sc1 = p1 + mask[s + 1];

        float mn   = fmaxf(m, fmaxf(sc0, sc1));
        float corr = __expf(m - mn);        // first iter: exp(-inf)=0
        float w0   = __expf(sc0 - mn);
        float w1   = __expf(sc1 - mn);
        l = l * corr + w0 + w1;
        acc.x = acc.x * corr + w0 * v0.x + w1 * v1.x;
        acc.y = acc.y * corr + w0 * v0.y + w1 * v1.y;
        acc.z = acc.z * corr + w0 * v0.z + w1 * v1.z;
        acc.w = acc.w * corr + w0 * v0.w + w1 * v1.w;
        m = mn;
    }
    if (s < s1 && s == MAX_CACHE) {          // the freshly-projected token
        const f4 kv = *(const f4*)(newK + (size_t)b * D_MODEL + headoff);
        const f4 vv = *(const f4*)(newV + (size_t)b * D_MODEL + headoff);
        float sc   = wave_allreduce_sum(dot4(qv, kv)) + mask[MAX_CACHE];
        float mn   = fmaxf(m, sc);
        float corr = __expf(m - mn);
        float w    = __expf(sc - mn);
        l = l * corr + w;
        acc.x = acc.x * corr + w * vv.x;
        acc.y = acc.y * corr + w * vv.y;
        acc.z = acc.z * corr + w * vv.z;
        acc.w = acc.w * corr + w * vv.w;
        m = mn;
    }

    // merge 4 wave states via LDS
    *(f4*)(&smem[wave * PART_STRIDE + dof]) = acc;
    if (lane == 0) {
        smem[wave * PART_STRIDE + 128] = m;
        smem[wave * PART_STRIDE + 129] = l;
    }
    __syncthreads();

    if (wave == 0) {
        float M = -INFINITY;
#pragma unroll
        for (int w = 0; w < 4; ++w)
            M = fmaxf(M, smem[w * PART_STRIDE + 128]);
        float L = 0.0f;
        f4 a = {0.0f, 0.0f, 0.0f, 0.0f};
#pragma unroll
        for (int w = 0; w < 4; ++w) {
            float scl = __expf(smem[w * PART_STRIDE + 128] - M);
            L += scl * smem[w * PART_STRIDE + 129];
            f4 t = *(const f4*)(&smem[w * PART_STRIDE + dof]);
            a.x += scl * t.x;
            a.y += scl * t.y;
            a.z += scl * t.z;
            a.w += scl * t.w;
        }
        float* pb = part + ((size_t)bh * NCHUNK + chunk) * PART_STRIDE;
        *(f4*)(pb + dof) = a;
        if (lane == 0) { pb[128] = M; pb[129] = L; }
    }
}

// ---------------------------------------------------------------------------
// Combine the NCHUNK partials of one (b,h) -> o_ws[b, h*128 + d].
// ---------------------------------------------------------------------------
__global__ __launch_bounds__(32)
void attn_combine(const float* __restrict__ part,
                  float* __restrict__ o)
{
    const int bh   = blockIdx.x;
    const int h    = bh % NUM_HEADS;
    const int b    = bh / NUM_HEADS;
    const int lane = threadIdx.x;
    const int dof  = lane * 4;

    const float* pb0 = part + (size_t)bh * NCHUNK * PART_STRIDE;

    float M = -INFINITY;
#pragma unroll
    for (int c = 0; c < NCHUNK; ++c)
        M = fmaxf(M, pb0[c * PART_STRIDE + 128]);

    float L = 0.0f;
    f4 acc = {0.0f, 0.0f, 0.0f, 0.0f};
#pragma unroll 4
    for (int c = 0; c < NCHUNK; ++c) {
        const float* pb = pb0 + c * PART_STRIDE;
        float sc = __expf(pb[128] - M);
        L += sc * pb[129];
        f4 a = *(const f4*)(pb + dof);
        acc.x += sc * a.x;
        acc.y += sc * a.y;
        acc.z += sc * a.z;
        acc.w += sc * a.w;
    }
    const float inv = 1.0f / L;
    float* op = o + (size_t)b * D_MODEL + h * HEAD_DIM + dof;
    f4 r = { acc.x * inv, acc.y * inv, acc.z * inv, acc.w * inv };
    *(f4*)op = r;
}

// ---------------------------------------------------------------------------
extern "C" void kernel_launch(void* const* d_in, const int* in_sizes, int n_in,
                              void* d_out, int out_size, void* d_ws, size_t ws_size,
                              hipStream_t stream)
{
    const float* hs   = (const float*)d_in[0];
    const float* pk   = (const float*)d_in[1];
    const float* pv   = (const float*)d_in[2];
    const float* mask = (const float*)d_in[3];
    const float* Wq   = (const float*)d_in[4];
    const float* bq   = (const float*)d_in[5];
    const float* Wk   = (const float*)d_in[6];
    const float* bk   = (const float*)d_in[7];
    const float* Wv   = (const float*)d_in[8];
    const float* bv   = (const float*)d_in[9];
    const float* Wo   = (const float*)d_in[10];
    const float* bo   = (const float*)d_in[11];

    float* out      = (float*)d_out;
    float* attn_out = out;                 // [16,1,2048]
    float* newK     = out + 32768;
    float* newV     = out + 65536;

    float* ws    = (float*)d_ws;
    float* q_ws  = ws;                     // 32768
    float* o_ws  = ws + 32768;             // 32768
    float* gpart = ws + 65536;             // 3*128*16*256 = 1,572,864 floats
    float* apart = gpart + 3 * NTILE * KSPLIT * 256;  // 16*16*17*132 floats

    // Fused Q/K/V projection: 6144 partial-tile waves, then 384-wave reduce.
    gemm16_wmma_partial<<<3 * NTILE * KSPLIT, 32, 0, stream>>>(hs, Wq, Wk, Wv, gpart);
    gemm16_reduce<<<3 * NTILE, 32, 0, stream>>>(gpart, bq, bk, bv,
                                                q_ws, newK, newV, SCALING);

    // Flash-decode over the 1 GB KV stream: 4352 blocks x 4 waves.
    attn_partial<<<BATCH * NUM_HEADS * NCHUNK, 128, 0, stream>>>(
        pk, pv, newK, newV, mask, q_ws, apart);
    attn_combine<<<BATCH * NUM_HEADS, 32, 0, stream>>>(apart, o_ws);

    // Output projection (proj index decodes to 0 for the whole grid).
    gemm16_wmma_partial<<<NTILE * KSPLIT, 32, 0, stream>>>(o_ws, Wo, Wo, Wo, gpart);
    gemm16_reduce<<<NTILE, 32, 0, stream>>>(gpart, bo, bo, bo,
                                            attn_out, attn_out, attn_out, 1.0f);
}